// NeuronClass_34548716929196
// MI455X (gfx1250) — compile-verified
//
#include <hip/hip_runtime.h>

// LIF neuron: drive = x @ w  (GEMV, memory-bound: 524 MB of x @ 23.3 TB/s ~= 22us floor),
// then a sequential LIF scan over T per batch row.
// Phase 1: V_WMMA_F32_16X16X4_F32 with w broadcast across the 16 B-columns.
// Data path: double-buffered GLOBAL_LOAD_ASYNC_TO_LDS_B128 (ASYNCcnt) -> ds_load_b64 frags,
// so 16 x 512B loads stay in flight per wave while WMMAs run on the other buffer.

typedef __attribute__((ext_vector_type(2))) float v2f;
typedef __attribute__((ext_vector_type(8))) float v8f;
typedef __attribute__((ext_vector_type(4))) int   v4i;

#define CHUNK 128            // K-elements staged per LDS pass (8 chunks for N=1024)
#define ROWPAD 132           // 128 + 4 floats pad -> frag reads hit all 64 banks once

#define AS1 __attribute__((address_space(1)))
#define AS3 __attribute__((address_space(3)))

#if defined(__has_builtin)
#if __has_builtin(__builtin_amdgcn_global_load_async_to_lds_b128)
#define HAVE_ASYNC_LDS 1
#endif
#if __has_builtin(__builtin_amdgcn_s_wait_asynccnt)
#define HAVE_WAIT_ASYNC 1
#endif
#endif

__device__ __forceinline__ void wait_asynccnt_0() {
#if defined(HAVE_WAIT_ASYNC)
    __builtin_amdgcn_s_wait_asynccnt(0);
#else
    asm volatile("s_wait_asynccnt 0x0" ::: "memory");
#endif
}
__device__ __forceinline__ void wait_asynccnt_16() {
#if defined(HAVE_WAIT_ASYNC)
    __builtin_amdgcn_s_wait_asynccnt(16);
#else
    asm volatile("s_wait_asynccnt 0x10" ::: "memory");
#endif
}

__global__ void lif_drive_wmma(const float* __restrict__ x,
                               const float* __restrict__ w,
                               float* __restrict__ drive,
                               int N, int numTiles) {
    __shared__ float s_x[2][16 * ROWPAD];   // double-buffered 16-row x CHUNK stage
    __shared__ float s_w[1024];             // full weight vector

    const int lane = threadIdx.x & 31;
    const int tile = blockIdx.x;
    if (tile >= numTiles) return;

    // preload w into LDS (coalesced b128)
    for (int i = lane; i < (N >> 2); i += 32) {
        ((float4*)s_w)[i] = ((const float4*)w)[i];
    }

    const int base = tile * 16;                 // first of 16 (b,t) rows
    const int h2   = (lane >> 4) << 1;          // 0 for lanes 0-15, 2 for lanes 16-31
    const int mrow = lane & 15;                 // A-matrix row owned by this lane
    const int NCH  = N / CHUNK;

    v8f acc = {0.f, 0.f, 0.f, 0.f, 0.f, 0.f, 0.f, 0.f};

#if defined(HAVE_ASYNC_LDS)
    // ---- async DMA path: global -> LDS without a VGPR round trip ----
    #define ISSUE_CHUNK(c0, bufi)                                                   \
        _Pragma("unroll")                                                           \
        for (int r = 0; r < 16; ++r) {                                              \
            __builtin_amdgcn_global_load_async_to_lds_b128(                         \
                (AS1 v4i*)(void*)&x[(size_t)(base + r) * N + (c0) + (lane << 2)],   \
                (AS3 v4i*)(void*)&s_x[bufi][r * ROWPAD + (lane << 2)], 0, 0);       \
        }

    ISSUE_CHUNK(0, 0)                                   // prologue: chunk 0 in flight
    for (int ci = 0; ci < NCH; ++ci) {
        const int buf = ci & 1;
        const int c0 = ci * CHUNK;
        if (ci + 1 < NCH) {                             // prefetch next chunk
            ISSUE_CHUNK(c0 + CHUNK, (ci + 1) & 1)
            wait_asynccnt_16();                         // in-order: chunk ci landed
        } else {
            wait_asynccnt_0();
        }
        asm volatile("" ::: "memory");                  // compiler fence vs. LDS reads
        asm volatile("s_wait_dscnt 0" ::: "memory");    // w-preload visible (1st iter)

        #pragma unroll
        for (int k = 0; k < CHUNK; k += 4) {
            v2f a = *(const v2f*)&s_x[buf][mrow * ROWPAD + k + h2]; // ds_load_b64
            v2f b = *(const v2f*)&s_w[c0 + k + h2];                 // broadcast b64
            acc = __builtin_amdgcn_wmma_f32_16x16x4_f32(
                false, a, false, b, (short)0, acc, false, false);
        }
        asm volatile("" ::: "memory");
    }
    #undef ISSUE_CHUNK
#else
    // ---- fallback: batch all 16 loads into regs, then store (keeps MLP) ----
    for (int ci = 0; ci < NCH; ++ci) {
        const int c0 = ci * CHUNK;
        float4 st[16];
        #pragma unroll
        for (int r = 0; r < 16; ++r)
            st[r] = *(const float4*)&x[(size_t)(base + r) * N + c0 + (lane << 2)];
        #pragma unroll
        for (int r = 0; r < 16; ++r)
            *(float4*)&s_x[0][r * ROWPAD + (lane << 2)] = st[r];
        asm volatile("s_wait_dscnt 0" ::: "memory");
        #pragma unroll
        for (int k = 0; k < CHUNK; k += 4) {
            v2f a = *(const v2f*)&s_x[0][mrow * ROWPAD + k + h2];
            v2f b = *(const v2f*)&s_w[c0 + k + h2];
            acc = __builtin_amdgcn_wmma_f32_16x16x4_f32(
                false, a, false, b, (short)0, acc, false, false);
        }
        asm volatile("s_wait_dscnt 0" ::: "memory");
    }
#endif

    // Every B-column holds the same dot product: lane 0 owns rows 0-7 in C VGPRs
    // 0-7, lane 16 owns rows 8-15.
    if ((lane & 15) == 0) {
        float* dv = drive + base + ((lane >> 4) << 3);
        dv[0] = acc[0]; dv[1] = acc[1]; dv[2] = acc[2]; dv[3] = acc[3];
        dv[4] = acc[4]; dv[5] = acc[5]; dv[6] = acc[6]; dv[7] = acc[7];
    }
}

__global__ void lif_scan(const float* __restrict__ drive,
                         float* __restrict__ out, int B, int T) {
    const int b = blockIdx.x * blockDim.x + threadIdx.x;
    if (b >= B) return;

    const float ALPHA = (float)(1.0 - 0.05 / 10.0);   // 0.995
    const float VTH   = 2.0f;

    const float4* d4 = (const float4*)(drive + (size_t)b * T);
    float4* v4 = (float4*)(out + (size_t)b * T);
    float4* z4 = (float4*)(out + (size_t)B * T + (size_t)b * T);

    float v = 0.f, z = 0.f;
    for (int i = 0; i < (T >> 2); ++i) {
        float4 di = d4[i];               // independent b128 load, pipelines ahead
        float4 vv, zz;
        v = ALPHA * v + di.x - VTH * z;  z = (v - VTH > 0.f) ? 1.f : 0.f; vv.x = v; zz.x = z;
        v = ALPHA * v + di.y - VTH * z;  z = (v - VTH > 0.f) ? 1.f : 0.f; vv.y = v; zz.y = z;
        v = ALPHA * v + di.z - VTH * z;  z = (v - VTH > 0.f) ? 1.f : 0.f; vv.z = v; zz.z = z;
        v = ALPHA * v + di.w - VTH * z;  z = (v - VTH > 0.f) ? 1.f : 0.f; vv.w = v; zz.w = z;
        v4[i] = vv;                      // coalesced-per-row b128 stores
        z4[i] = zz;
    }
}

extern "C" void kernel_launch(void* const* d_in, const int* in_sizes, int n_in,
                              void* d_out, int out_size, void* d_ws, size_t ws_size,
                              hipStream_t stream) {
    const float* x = (const float*)d_in[0];   // [B, T, N] fp32
    const float* w = (const float*)d_in[1];   // [N] fp32
    float* out = (float*)d_out;               // [2, B, T] fp32 (v_seq then z_seq)

    const int N = in_sizes[1];                              // 1024
    const int R = (int)((long long)in_sizes[0] / N);        // B*T = 128000
    const int T = 1000;
    const int B = R / T;                                    // 128

    float* drive = (float*)d_ws;              // R floats of scratch

    const int numTiles = R / 16;              // 8000 tiles of 16 rows
    lif_drive_wmma<<<numTiles, 32, 0, stream>>>(x, w, drive, N, numTiles);
    lif_scan<<<(B + 127) / 128, 128, 0, stream>>>(drive, out, B, T);
}